// HierarchicalBG_60155311948495
// MI455X (gfx1250) — compile-verified
//
#include <hip/hip_runtime.h>
#include <hip/hip_bf16.h>
#include <math.h>

// ---------------------------------------------------------------------------
// HierarchicalBG: pano bicubic gather (8ch, 3 levels) + tiny MLP (8->128->3)
// Gather: 1 ray/lane, fast-path 16B row loads. MLP GEMM1 via
// V_WMMA_F32_16X16X4_F32 (f32, matches reference precision), GEMM2 via
// per-lane partials + wave32 shfl_xor reduction.
// ---------------------------------------------------------------------------

typedef float v2f __attribute__((ext_vector_type(2)));
typedef float v8f __attribute__((ext_vector_type(8)));

#define RANK    8
#define FEATC   128
#define NBLKTH  256   // 8 waves of 32

__device__ __forceinline__ void cubic_w(float t, float w[4]) {
  // A = -0.75
  float t1 = 1.0f + t;
  float t2 = 2.0f - t;
  float u  = 1.0f - t;
  w[0] = ((-0.75f * t1 + 3.75f) * t1 - 6.0f) * t1 + 3.0f;  // c2(1+t)
  w[1] = (1.25f * t - 2.25f) * t * t + 1.0f;               // c1(t)
  w[2] = (1.25f * u - 2.25f) * u * u + 1.0f;               // c1(1-t)
  w[3] = ((-0.75f * t2 + 3.75f) * t2 - 6.0f) * t2 + 3.0f;  // c2(2-t)
}

__device__ __forceinline__ void bicubic_accum(const float* __restrict__ img,
                                              int H, int W,
                                              float gx, float gy, float scale,
                                              float emb[RANK]) {
  float ix = ((gx + 1.0f) * (float)W - 1.0f) * 0.5f;
  float iy = ((gy + 1.0f) * (float)H - 1.0f) * 0.5f;
  float fx = floorf(ix), fy = floorf(iy);
  float wx[4], wy[4];
  cubic_w(ix - fx, wx);
  cubic_w(iy - fy, wy);
  int x0 = (int)fx, y0 = (int)fy;
  const size_t HW = (size_t)H * (size_t)W;

  float acc[RANK];
#pragma unroll
  for (int c = 0; c < RANK; ++c) acc[c] = 0.0f;

  const bool xfast = (x0 >= 1) && (x0 + 2 < W);

#pragma unroll
  for (int dy = 0; dy < 4; ++dy) {
    int yy = y0 - 1 + dy;
    if (yy < 0 || yy >= H) continue;    // row masked to zero in reference
    float wrow = wy[dy];
    size_t rowbase = (size_t)yy * (size_t)W;
    if (xfast) {
      size_t base = rowbase + (size_t)(x0 - 1);
#pragma unroll
      for (int c = 0; c < RANK; ++c) {
        float v4[4];
        __builtin_memcpy(v4, img + (size_t)c * HW + base, 16);  // b128, 4B-aligned ok
        acc[c] += wrow * (wx[0] * v4[0] + wx[1] * v4[1] + wx[2] * v4[2] + wx[3] * v4[3]);
      }
    } else {
#pragma unroll
      for (int dx = 0; dx < 4; ++dx) {
        int xx = x0 - 1 + dx;
        if (xx < 0 || xx >= W) continue;  // column masked to zero
        float wv = wrow * wx[dx];
#pragma unroll
        for (int c = 0; c < RANK; ++c)
          acc[c] += wv * img[(size_t)c * HW + rowbase + (size_t)xx];
      }
    }
  }
#pragma unroll
  for (int c = 0; c < RANK; ++c) emb[c] += scale * acc[c];
}

__global__ __launch_bounds__(NBLKTH) void hierbg_kernel(
    const float* __restrict__ vdirs,     // (B,3)
    const float* __restrict__ saSample,  // (B,)
    const float* __restrict__ m0,        // (8, 512, 1024)
    const float* __restrict__ m1,        // (8, 1024, 2048)
    const float* __restrict__ m2,        // (8, 2048, 4096)
    const float* __restrict__ W1,        // (128, 8) row-major
    const float* __restrict__ W2,        // (3, 128) row-major
    float* __restrict__ out)             // (B, 3)
{
  const int tid = threadIdx.x;
  const int ray = blockIdx.x * NBLKTH + tid;

  // ---- unwrap direction -> (gx, gy) ----
  float dx = vdirs[ray * 3 + 0];
  float dyv = vdirs[ray * 3 + 1];
  float dz = vdirs[ray * 3 + 2];
  const float INV_PI = 0.3183098861837907f;
  float gx = atan2f(dx, dz) * INV_PI;
  float gy = acosf(fminf(fmaxf(dyv, -1.0f), 1.0f)) * (2.0f * INV_PI) - 1.0f;

  // ---- mip from solid angle (res = H of finest level = 2048) ----
  const float saTexel = 4.0f * 3.14159265358979323846f / (6.0f * 2048.0f * 2048.0f);
  float mip = __logf(saSample[ray] / saTexel) * (1.0f / (2.0f * 0.69314718055994531f));
  mip = fminf(fmaxf(mip, 1.0f), 3.0f);

  // ---- hierarchical bicubic gather into emb[8] ----
  float emb[RANK];
#pragma unroll
  for (int c = 0; c < RANK; ++c) emb[c] = 0.0f;

  {
    float w0 = fminf(fmaxf(3.0f - mip, 0.0f), 1.0f);
    if (w0 > 0.0f) bicubic_accum(m0, 512, 1024, gx, gy, w0, emb);
    float w1 = fminf(fmaxf(2.0f - mip, 0.0f), 1.0f);
    if (w1 > 0.0f) bicubic_accum(m1, 1024, 2048, gx, gy, w1 * 0.5f, emb);
    float w2 = fminf(fmaxf(1.0f - mip, 0.0f), 1.0f);   // always 0 (mip>=1); kept for fidelity
    if (w2 > 0.0f) bicubic_accum(m2, 2048, 4096, gx, gy, w2 * (1.0f / 3.0f), emb);
  }

  // ---- stage emb through LDS to build WMMA A-operands ----
  __shared__ float semb[NBLKTH][RANK + 1];  // +1 pad: avoid 8-way bank conflict
#pragma unroll
  for (int c = 0; c < RANK; ++c) semb[tid][c] = emb[c];
  __syncthreads();

  const int wave = tid >> 5;
  const int lane = tid & 31;
  const int lhalf = lane >> 4;   // 0: K={0,1}/M rows 0-7 of C, 1: K={2,3}/M rows 8-15
  const int ln = lane & 15;

  // B operands for GEMM1 (shared across both ray tiles):
  // B[k][n] = W1[nt*16 + n][k], striping: vgpr v, lane-half h -> K = kc*4 + 2h + v
  v2f bop[8][2];
#pragma unroll
  for (int nt = 0; nt < 8; ++nt) {
#pragma unroll
    for (int kc = 0; kc < 2; ++kc) {
      int f = nt * 16 + ln;
      int k = kc * 4 + lhalf * 2;
      v2f b;
      b.x = W1[f * RANK + k];
      b.y = W1[f * RANK + k + 1];
      bop[nt][kc] = b;
    }
  }

  const int blockRayBase = blockIdx.x * NBLKTH + wave * 32;

#pragma unroll
  for (int t = 0; t < 2; ++t) {           // two 16-ray M-tiles per wave
    const int rloc = wave * 32 + t * 16 + ln;

    float p[8][3];                        // GEMM2 per-lane partials
#pragma unroll
    for (int j = 0; j < 8; ++j) {
      p[j][0] = 0.0f; p[j][1] = 0.0f; p[j][2] = 0.0f;
    }

#pragma unroll
    for (int nt = 0; nt < 8; ++nt) {      // 8 N-tiles cover FEATC=128
      v8f cacc = {};
#pragma unroll
      for (int kc = 0; kc < 2; ++kc) {    // K=8 in two chunks of 4
        int k = kc * 4 + lhalf * 2;
        v2f a;
        a.x = semb[rloc][k];
        a.y = semb[rloc][k + 1];
        // D = A(16x4,f32) * B(4x16,f32) + C(16x16,f32)
        cacc = __builtin_amdgcn_wmma_f32_16x16x4_f32(
            false, a, false, bop[nt][kc], (short)0, cacc, false, false);
      }
      // relu(h) and fold into GEMM2 partials.
      // C layout: c[j] = h[M = j + 8*lhalf][N = ln] (feature nt*16+ln)
      float w2c0 = W2[0 * FEATC + nt * 16 + ln];
      float w2c1 = W2[1 * FEATC + nt * 16 + ln];
      float w2c2 = W2[2 * FEATC + nt * 16 + ln];
#pragma unroll
      for (int j = 0; j < 8; ++j) {
        float h = fmaxf(cacc[j], 0.0f);
        p[j][0] += h * w2c0;
        p[j][1] += h * w2c1;
        p[j][2] += h * w2c2;
      }
    }

    // Reduce partials across the 16 lanes of each half (wave32 shfl).
#pragma unroll
    for (int j = 0; j < 8; ++j) {
#pragma unroll
      for (int ch = 0; ch < 3; ++ch) {
        float v = p[j][ch];
        v += __shfl_xor(v, 1);
        v += __shfl_xor(v, 2);
        v += __shfl_xor(v, 4);
        v += __shfl_xor(v, 8);
        p[j][ch] = v;
      }
    }

    if (ln == 0) {                        // lane 0 -> rows 0-7, lane 16 -> rows 8-15
#pragma unroll
      for (int j = 0; j < 8; ++j) {
        int m = j + 8 * lhalf;
        int rg = blockRayBase + t * 16 + m;
#pragma unroll
        for (int ch = 0; ch < 3; ++ch)
          out[rg * 3 + ch] = p[j][ch];
      }
    }
  }
}

extern "C" void kernel_launch(void* const* d_in, const int* in_sizes, int n_in,
                              void* d_out, int out_size, void* d_ws, size_t ws_size,
                              hipStream_t stream) {
  const float* vdirs = (const float*)d_in[0];
  const float* sa    = (const float*)d_in[1];
  const float* m0    = (const float*)d_in[2];
  const float* m1    = (const float*)d_in[3];
  const float* m2    = (const float*)d_in[4];
  const float* W1    = (const float*)d_in[5];
  const float* W2    = (const float*)d_in[6];
  float* out = (float*)d_out;

  const int B = in_sizes[0] / 3;          // 262144
  const int blocks = B / NBLKTH;          // 1024
  hierbg_kernel<<<blocks, NBLKTH, 0, stream>>>(vdirs, sa, m0, m1, m2, W1, W2, out);
}